// Hilbert_2DSelectiveScan_26207890440780
// MI455X (gfx1250) — compile-verified
//
#include <hip/hip_runtime.h>
#include <hip/hip_bf16.h>
#include <math.h>

// ---------------- problem constants ----------------
#define Bn      4
#define Hn      64
#define Wn      64
#define Ln      4096            // H*W
#define BLn     16384           // B*L
#define DM      192             // d_model
#define DI      384             // d_inner
#define DS      16              // d_state
#define DTR     12              // dt_rank
#define Kd      2
#define NSEG    32
#define SEGLEN  128             // NSEG*SEGLEN == L
#define LN_EPS  1e-5f
#define PF_DIST 8               // scan prefetch distance (steps)

typedef __attribute__((ext_vector_type(2))) float v2f;
typedef __attribute__((ext_vector_type(8))) float v8f;

// V_WMMA_F32_16X16X4_F32 : D(16x16,f32) = A(16x4,f32) x B(4x16,f32) + C
// A frag: lane m=lane&15 holds K = 2*(lane>>4) + {0,1}  (2 VGPRs)
// B frag: lane n=lane&15 holds K = 2*(lane>>4) + {0,1}  (2 VGPRs)
// C/D   : VGPR v, lane -> M = v + 8*(lane>>4), N = lane&15
__device__ __forceinline__ v8f wmma4(v2f a, v2f b, v8f c) {
    return __builtin_amdgcn_wmma_f32_16x16x4_f32(
        false, a, false, b, (short)0, c, false, false);
}

__device__ __forceinline__ float fast_silu(float x) {
    return x / (1.0f + __expf(-x));
}
__device__ __forceinline__ float softplusf(float x) {
    return (x > 20.0f) ? x : log1pf(__expf(x));
}

// ---------------- K0: inverse Hilbert permutation ----------------
__global__ void build_inv_kernel(const int* __restrict__ hil, int* __restrict__ inv) {
    int i = blockIdx.x * 256 + threadIdx.x;
    if (i < Ln) inv[hil[i]] = i;
}

// ---------------- K1: xz = x @ W_in.T  (16384x192 * 192x768) ----------------
// block = 384 thr (12 waves, one N-group of 64 cols each); 64 rows per block.
// A tile (64x192) staged in LDS (row stride 196 -> bank-conflict-free frags).
#define S1  196
__global__ void __launch_bounds__(384)
gemm_in_kernel(const float* __restrict__ X, const float* __restrict__ Win,
               float* __restrict__ x1, float* __restrict__ z) {
    __shared__ float sA[64 * S1];
    int tid  = threadIdx.x;
    int rowbase = blockIdx.x * 64;
    // cooperative stage: 64*192 floats, 32 per thread
    {
        int c = tid % 192, rh = tid / 192;
        #pragma unroll
        for (int i = 0; i < 32; i++) {
            int r = i * 2 + rh;
            sA[r * S1 + c] = X[(size_t)(rowbase + r) * DM + c];
        }
    }
    __syncthreads();

    int ng   = tid >> 5;          // N-group 0..11 (64 cols each)
    int lane = tid & 31;
    int koff = (lane >> 4) << 1;
    int nb   = ng * 64 + (lane & 15);

    v8f acc[4][4] = {};
    for (int k0 = 0; k0 < DM; k0 += 4) {
        v2f b0 = *(const v2f*)(Win + (size_t)(nb +  0) * DM + k0 + koff);
        v2f b1 = *(const v2f*)(Win + (size_t)(nb + 16) * DM + k0 + koff);
        v2f b2 = *(const v2f*)(Win + (size_t)(nb + 32) * DM + k0 + koff);
        v2f b3 = *(const v2f*)(Win + (size_t)(nb + 48) * DM + k0 + koff);
        #pragma unroll
        for (int mt = 0; mt < 4; mt++) {
            v2f a = *(const v2f*)&sA[(mt * 16 + (lane & 15)) * S1 + k0 + koff];
            acc[mt][0] = wmma4(a, b0, acc[mt][0]);
            acc[mt][1] = wmma4(a, b1, acc[mt][1]);
            acc[mt][2] = wmma4(a, b2, acc[mt][2]);
            acc[mt][3] = wmma4(a, b3, acc[mt][3]);
        }
    }
    int mbase = (lane >> 4) << 3;
    #pragma unroll
    for (int mt = 0; mt < 4; mt++) {
        #pragma unroll
        for (int t = 0; t < 4; t++) {
            int col = ng * 64 + t * 16 + (lane & 15);
            #pragma unroll
            for (int v = 0; v < 8; v++) {
                size_t row = (size_t)(rowbase + mt * 16 + mbase + v);
                float val = acc[mt][t][v];
                if (col < DI) x1[row * DI + col] = val;
                else          z [row * DI + (col - DI)] = val;
            }
        }
    }
}

// ---------------- K2: depthwise 3x3 SAME conv + SiLU + Hilbert gather ----------------
__global__ void conv_silu_gather_kernel(const float* __restrict__ x1,
                                        const float* __restrict__ cw,
                                        const float* __restrict__ cb,
                                        const int* __restrict__ inv,
                                        float* __restrict__ seq) {
    int pg = blockIdx.x;            // b*4096 + p (raster)
    int d  = threadIdx.x;           // channel
    int b  = pg >> 12, p = pg & (Ln - 1);
    int h  = p >> 6,  w = p & 63;
    float acc = cb[d];
    #pragma unroll
    for (int kh = 0; kh < 3; kh++) {
        int hh = h + kh - 1;
        if (hh < 0 || hh > 63) continue;
        #pragma unroll
        for (int kw = 0; kw < 3; kw++) {
            int ww = w + kw - 1;
            if (ww < 0 || ww > 63) continue;
            acc += x1[((size_t)b * Ln + hh * 64 + ww) * DI + d] * cw[d * 9 + kh * 3 + kw];
        }
    }
    seq[((size_t)b * Ln + inv[p]) * DI + d] = fast_silu(acc);
}

// ---------------- K3: x_dbl = seq @ x_proj_weight.T for both k ----------------
// block = 192 thr (6 waves, one (k, c-tile) each); 32 rows per block.
#define S2  388
__global__ void __launch_bounds__(192)
gemm_xproj_kernel(const float* __restrict__ seq, const float* __restrict__ xpw,
                  float* __restrict__ dbl) {
    __shared__ float sA[32 * S2];
    int tid = threadIdx.x;
    int rowbase = blockIdx.x * 32;
    #pragma unroll
    for (int i = 0; i < 32; i++) {
        sA[i * S2 + tid]       = seq[(size_t)(rowbase + i) * DI + tid];
        sA[i * S2 + 192 + tid] = seq[(size_t)(rowbase + i) * DI + 192 + tid];
    }
    __syncthreads();

    int wave = tid >> 5;           // 0..5
    int kt = wave / 3, ct = wave % 3;
    int lane = tid & 31;
    int koff = (lane >> 4) << 1;
    int cc = ct * 16 + (lane & 15);

    v8f acc0 = {}, acc1 = {};
    for (int k0 = 0; k0 < DI; k0 += 4) {
        v2f bf = {0.0f, 0.0f};
        if (cc < 44)
            bf = *(const v2f*)(xpw + (size_t)(kt * 44 + cc) * DI + k0 + koff);
        v2f a0 = *(const v2f*)&sA[((lane & 15)) * S2 + k0 + koff];
        v2f a1 = *(const v2f*)&sA[(16 + (lane & 15)) * S2 + k0 + koff];
        acc0 = wmma4(a0, bf, acc0);
        acc1 = wmma4(a1, bf, acc1);
    }
    if (cc < 44) {
        int mbase = (lane >> 4) << 3;
        #pragma unroll
        for (int v = 0; v < 8; v++) {
            size_t r0 = (size_t)(rowbase + mbase + v);
            size_t r1 = (size_t)(rowbase + 16 + mbase + v);
            dbl[((size_t)kt * BLn + r0) * 48 + cc] = acc0[v];
            dbl[((size_t)kt * BLn + r1) * 48 + cc] = acc1[v];
        }
    }
}

// ---------------- chunked selective scan ----------------
// chan = (b*2+k)*384 + d ; segments in scan-index space.
// dbl row: [0..11]=dt_raw, [12..27]=B, [28..43]=C (indexed by seq position).
__global__ void scan_phaseA_kernel(const float* __restrict__ seq, const float* __restrict__ dbl,
                                   const float* __restrict__ Alogs, const float* __restrict__ dtw_g,
                                   const float* __restrict__ dtb,
                                   float* __restrict__ Pseg, float* __restrict__ Gseg) {
    int seg  = blockIdx.x & (NSEG - 1);
    int t    = blockIdx.x >> 5;
    int dblk = t % 12;
    int bk   = t / 12;              // b*2+k
    int k = bk & 1, b = bk >> 1;
    int d  = dblk * 32 + threadIdx.x;
    int kd = k * DI + d;

    float Av[DS];
    #pragma unroll
    for (int n = 0; n < DS; n++) Av[n] = -__expf(Alogs[(size_t)kd * DS + n]);
    float dw[DTR];
    #pragma unroll
    for (int r = 0; r < DTR; r++) dw[r] = dtw_g[(size_t)kd * DTR + r];
    float bias = dtb[kd];

    float h[DS], P[DS];
    #pragma unroll
    for (int n = 0; n < DS; n++) { h[n] = 0.0f; P[n] = 1.0f; }

    const float* dblb = dbl + ((size_t)k * BLn + (size_t)b * Ln) * 48;
    const float* seqb = seq + (size_t)b * Ln * DI;

    for (int j = 0; j < SEGLEN; j++) {
        int m   = seg * SEGLEN + j;
        int pos = k ? (Ln - 1 - m) : m;       // seq position consumed
        // prefetch a few steps ahead (global_prefetch_b8)
        int jp = (j + PF_DIST < SEGLEN) ? (j + PF_DIST) : (SEGLEN - 1);
        int mp = seg * SEGLEN + jp;
        int posp = k ? (Ln - 1 - mp) : mp;
        __builtin_prefetch(dblb + (size_t)posp * 48, 0, 3);
        __builtin_prefetch(seqb + (size_t)posp * DI + d, 0, 3);

        const float* row = dblb + (size_t)pos * 48;
        float din = bias;
        #pragma unroll
        for (int r = 0; r < DTR; r++) din = fmaf(row[r], dw[r], din);
        float delta = softplusf(din);
        float du = delta * seqb[(size_t)pos * DI + d];
        #pragma unroll
        for (int n = 0; n < DS; n++) {
            float dA = __expf(delta * Av[n]);
            P[n] *= dA;
            h[n] = fmaf(dA, h[n], du * row[12 + n]);
        }
    }
    int chan = bk * DI + d;
    size_t o = ((size_t)chan * NSEG + seg) * DS;
    #pragma unroll
    for (int n = 0; n < DS; n++) { Pseg[o + n] = P[n]; Gseg[o + n] = h[n]; }
}

__global__ void scan_phaseB_kernel(const float* __restrict__ Pseg, const float* __restrict__ Gseg,
                                   float* __restrict__ Hin) {
    int gid  = blockIdx.x * blockDim.x + threadIdx.x;   // 3072*16 items
    int chan = gid >> 4, n = gid & 15;
    float h = 0.0f;
    for (int s = 0; s < NSEG; s++) {
        size_t o = ((size_t)chan * NSEG + s) * DS + n;
        Hin[o] = h;
        h = fmaf(Pseg[o], h, Gseg[o]);
    }
}

__global__ void scan_phaseC_kernel(const float* __restrict__ seq, const float* __restrict__ dbl,
                                   const float* __restrict__ Alogs, const float* __restrict__ dtw_g,
                                   const float* __restrict__ dtb, const float* __restrict__ Hin,
                                   float* __restrict__ y0, float* __restrict__ y1) {
    int seg  = blockIdx.x & (NSEG - 1);
    int t    = blockIdx.x >> 5;
    int dblk = t % 12;
    int bk   = t / 12;
    int k = bk & 1, b = bk >> 1;
    int d  = dblk * 32 + threadIdx.x;
    int kd = k * DI + d;

    float Av[DS];
    #pragma unroll
    for (int n = 0; n < DS; n++) Av[n] = -__expf(Alogs[(size_t)kd * DS + n]);
    float dw[DTR];
    #pragma unroll
    for (int r = 0; r < DTR; r++) dw[r] = dtw_g[(size_t)kd * DTR + r];
    float bias = dtb[kd];

    int chan = bk * DI + d;
    size_t ho = ((size_t)chan * NSEG + seg) * DS;
    float h[DS];
    #pragma unroll
    for (int n = 0; n < DS; n++) h[n] = Hin[ho + n];

    const float* dblb = dbl + ((size_t)k * BLn + (size_t)b * Ln) * 48;
    const float* seqb = seq + (size_t)b * Ln * DI;
    float* yb = k ? y1 : y0;

    for (int j = 0; j < SEGLEN; j++) {
        int m   = seg * SEGLEN + j;
        int pos = k ? (Ln - 1 - m) : m;
        int jp = (j + PF_DIST < SEGLEN) ? (j + PF_DIST) : (SEGLEN - 1);
        int mp = seg * SEGLEN + jp;
        int posp = k ? (Ln - 1 - mp) : mp;
        __builtin_prefetch(dblb + (size_t)posp * 48, 0, 3);
        __builtin_prefetch(seqb + (size_t)posp * DI + d, 0, 3);

        const float* row = dblb + (size_t)pos * 48;
        float din = bias;
        #pragma unroll
        for (int r = 0; r < DTR; r++) din = fmaf(row[r], dw[r], din);
        float delta = softplusf(din);
        float du = delta * seqb[(size_t)pos * DI + d];
        float y = 0.0f;
        #pragma unroll
        for (int n = 0; n < DS; n++) {
            float dA = __expf(delta * Av[n]);
            h[n] = fmaf(dA, h[n], du * row[12 + n]);
            y = fmaf(h[n], row[28 + n], y);
        }
        yb[((size_t)b * Ln + pos) * DI + d] = y;   // indexed by seq position
    }
}

// ---------------- F1: combine + LayerNorm + SiLU gate ----------------
__global__ void ln_gate_kernel(const float* __restrict__ y0, const float* __restrict__ y1,
                               const float* __restrict__ seq, const float* __restrict__ z,
                               const float* __restrict__ Dsv, const float* __restrict__ gamma,
                               const float* __restrict__ beta, const int* __restrict__ hil,
                               float* __restrict__ yg) {
    int lg = blockIdx.x;                // b*4096 + l (seq index)
    int d  = threadIdx.x;               // 0..383
    int b  = lg >> 12, l = lg & (Ln - 1);
    size_t base = (size_t)lg * DI;
    float v = y0[base + d] + y1[base + d] + (Dsv[d] + Dsv[DI + d]) * seq[base + d];

    float s = v, q = v * v;
    #pragma unroll
    for (int off = 16; off > 0; off >>= 1) {
        s += __shfl_down(s, off);
        q += __shfl_down(q, off);
    }
    __shared__ float ss[12], sq[12];
    int wid = d >> 5, lane = d & 31;
    if (lane == 0) { ss[wid] = s; sq[wid] = q; }
    __syncthreads();
    if (d == 0) {
        float a = 0.0f, c = 0.0f;
        #pragma unroll
        for (int i = 0; i < 12; i++) { a += ss[i]; c += sq[i]; }
        ss[0] = a; sq[0] = c;
    }
    __syncthreads();
    float mu  = ss[0] * (1.0f / DI);
    float var = sq[0] * (1.0f / DI) - mu * mu;
    float yn  = (v - mu) * rsqrtf(var + LN_EPS) * gamma[d] + beta[d];

    int p = hil[l];
    size_t zi = ((size_t)b * Ln + p) * DI + d;
    yg[zi] = yn * fast_silu(z[zi]);
}

// ---------------- F2: out = yg @ W_out.T (16384x384 * 384x192) ----------------
// block = 384 thr (12 waves, one 16-col N-tile each); 32 rows per block.
__global__ void __launch_bounds__(384)
gemm_out_kernel(const float* __restrict__ yg, const float* __restrict__ Wout,
                float* __restrict__ out) {
    __shared__ float sA[32 * S2];
    int tid = threadIdx.x;
    int rowbase = blockIdx.x * 32;
    #pragma unroll
    for (int i = 0; i < 32; i++)
        sA[i * S2 + tid] = yg[(size_t)(rowbase + i) * DI + tid];
    __syncthreads();

    int nt   = tid >> 5;            // 0..11 (16 cols each)
    int lane = tid & 31;
    int koff = (lane >> 4) << 1;
    int n    = nt * 16 + (lane & 15);

    v8f acc0 = {}, acc1 = {};
    for (int k0 = 0; k0 < DI; k0 += 4) {
        v2f bf = *(const v2f*)(Wout + (size_t)n * DI + k0 + koff);
        v2f a0 = *(const v2f*)&sA[((lane & 15)) * S2 + k0 + koff];
        v2f a1 = *(const v2f*)&sA[(16 + (lane & 15)) * S2 + k0 + koff];
        acc0 = wmma4(a0, bf, acc0);
        acc1 = wmma4(a1, bf, acc1);
    }
    int mbase = (lane >> 4) << 3;
    #pragma unroll
    for (int v = 0; v < 8; v++) {
        out[(size_t)(rowbase + mbase + v) * DM + n]      = acc0[v];
        out[(size_t)(rowbase + 16 + mbase + v) * DM + n] = acc1[v];
    }
}

// ---------------- launcher ----------------
extern "C" void kernel_launch(void* const* d_in, const int* in_sizes, int n_in,
                              void* d_out, int out_size, void* d_ws, size_t ws_size,
                              hipStream_t stream) {
    const float* x     = (const float*)d_in[0];
    const int*   hil   = (const int*)  d_in[1];
    const float* Win   = (const float*)d_in[2];
    const float* convw = (const float*)d_in[3];
    const float* convb = (const float*)d_in[4];
    const float* xpw   = (const float*)d_in[5];
    const float* dtw   = (const float*)d_in[6];
    const float* dtb   = (const float*)d_in[7];
    const float* Alogs = (const float*)d_in[8];
    const float* Dsv   = (const float*)d_in[9];
    const float* gamma = (const float*)d_in[10];
    const float* beta  = (const float*)d_in[11];
    const float* Wout  = (const float*)d_in[12];
    float* out = (float*)d_out;

    char* ws = (char*)d_ws;
    size_t off = 0;
    auto alloc = [&](size_t bytes) { char* p = ws + off; off = (off + bytes + 255) & ~(size_t)255; return p; };
    float* x1    = (float*)alloc((size_t)BLn * DI * 4);
    float* zbuf  = (float*)alloc((size_t)BLn * DI * 4);
    float* seq   = (float*)alloc((size_t)BLn * DI * 4);
    float* dbl   = (float*)alloc((size_t)Kd * BLn * 48 * 4);
    float* Pseg  = (float*)alloc((size_t)3072 * NSEG * DS * 4);
    float* Gseg  = (float*)alloc((size_t)3072 * NSEG * DS * 4);
    float* Hin   = (float*)alloc((size_t)3072 * NSEG * DS * 4);
    float* y0    = (float*)alloc((size_t)BLn * DI * 4);
    float* y1    = (float*)alloc((size_t)BLn * DI * 4);
    float* yg    = (float*)alloc((size_t)BLn * DI * 4);
    int*   inv   = (int*)  alloc((size_t)Ln * 4);

    build_inv_kernel<<<dim3(Ln / 256), dim3(256), 0, stream>>>(hil, inv);
    gemm_in_kernel<<<dim3(BLn / 64), dim3(384), 0, stream>>>(x, Win, x1, zbuf);
    conv_silu_gather_kernel<<<dim3(BLn), dim3(DI), 0, stream>>>(x1, convw, convb, inv, seq);
    gemm_xproj_kernel<<<dim3(BLn / 32), dim3(192), 0, stream>>>(seq, xpw, dbl);
    scan_phaseA_kernel<<<dim3(Bn * Kd * 12 * NSEG), dim3(32), 0, stream>>>(
        seq, dbl, Alogs, dtw, dtb, Pseg, Gseg);
    scan_phaseB_kernel<<<dim3((3072 * DS) / 128), dim3(128), 0, stream>>>(Pseg, Gseg, Hin);
    scan_phaseC_kernel<<<dim3(Bn * Kd * 12 * NSEG), dim3(32), 0, stream>>>(
        seq, dbl, Alogs, dtw, dtb, Hin, y0, y1);
    ln_gate_kernel<<<dim3(BLn), dim3(DI), 0, stream>>>(
        y0, y1, seq, zbuf, Dsv, gamma, beta, hil, yg);
    gemm_out_kernel<<<dim3(BLn / 32), dim3(384), 0, stream>>>(yg, Wout, out);
}